// GNN_63247688401685
// MI455X (gfx1250) — compile-verified
//
#include <hip/hip_runtime.h>
#include <math.h>

// MI455X (gfx1250) implementation of the GNN transformer-conv layer.
//
// Phases:
//  1. gnn_init        : segmax=-inf, denom=0, column sums=0
//  2. gnn_gemm_qkvs   : Q,K,V = x@W+b via V_WMMA_F32_16X16X4_F32 (fp32 matrix pipe),
//                       skip term written directly into the ACC accumulator.
//  3. gnn_edge_logits : wave-per-edge coalesced q[dst]*k[src] dots (+ float atomic max)
//  4. gnn_edge_exp    : ex = exp(logit - segmax[dst]); denom += ex
//  5. gnn_scatter     : ACC[dst] += (ex/denom) * v[src]   (global_atomic_add_f32)
//  6. gnn_stats       : per-column sum / sum-of-squares over N
//  7. gnn_bn_mish     : BatchNorm (batch stats, biased var) + Mish -> d_out
//
// Workspace layout (floats): Q,K,V,ACC (N*128 each), EX (E*2), SEGMAX (N*2),
// DENOM (N*2), SUMS(128), SUMSQ(128)  ->  ~110 MB for N=50000, E=800000.

typedef float v2f __attribute__((ext_vector_type(2)));
typedef float v8f __attribute__((ext_vector_type(8)));

#define IN_DIM 128
#define EPS_BN 1e-5f

// ---------------------------------------------------------------- float atomic max
__device__ __forceinline__ void atomicMaxFloat(float* addr, float val) {
  // sign-split trick: int-max for non-negative, uint-min for negative
  if (val >= 0.0f) {
    atomicMax((int*)addr, __float_as_int(val));
  } else {
    atomicMin((unsigned int*)addr, __float_as_uint(val));
  }
}

// ---------------------------------------------------------------- 1. init
__global__ void gnn_init(float* __restrict__ segmax, float* __restrict__ denom,
                         int NH, float* __restrict__ sums, float* __restrict__ sumsq) {
  int tid = blockIdx.x * blockDim.x + threadIdx.x;
  if (tid < NH) {
    segmax[tid] = -__builtin_inff();
    denom[tid]  = 0.0f;
  }
  if (tid < IN_DIM) {
    sums[tid]  = 0.0f;
    sumsq[tid] = 0.0f;
  }
}

// ---------------------------------------------------------------- 2. WMMA GEMM
// grid = (ceil(N/16), 4): blockIdx.y selects {Q,K,V,skip}. 256 threads = 8 waves;
// wave w computes the 16x16 output tile at columns [16w, 16w+16).
__global__ __launch_bounds__(256) void gnn_gemm_qkvs(
    const float* __restrict__ x, int N,
    const float* __restrict__ Wq, const float* __restrict__ bq,
    const float* __restrict__ Wk, const float* __restrict__ bk,
    const float* __restrict__ Wv, const float* __restrict__ bv,
    const float* __restrict__ Ws, const float* __restrict__ bs,
    float* __restrict__ Q, float* __restrict__ K,
    float* __restrict__ V, float* __restrict__ ACC) {
  __shared__ float xt[16 * 132];  // 16x128 tile, stride 132 to avoid bank conflicts

  const int m = blockIdx.y;
  const float* W  = (m == 0) ? Wq : (m == 1) ? Wk : (m == 2) ? Wv : Ws;
  const float* bb = (m == 0) ? bq : (m == 1) ? bk : (m == 2) ? bv : bs;
  float*       O  = (m == 0) ? Q  : (m == 1) ? K  : (m == 2) ? V  : ACC;

  const int row0 = blockIdx.x * 16;
  const int t = threadIdx.x;

  // stage x[row0:row0+16, 0:128] into LDS (8 floats per thread)
  {
    int idx = t * 8;
    int r = idx >> 7;        // 0..15
    int c = idx & 127;       // 0,8,16,...
    const float* src = x + (size_t)(row0 + r) * IN_DIM + c;
    bool ok = (row0 + r) < N;
#pragma unroll
    for (int j = 0; j < 8; ++j) xt[r * 132 + c + j] = ok ? src[j] : 0.0f;
  }
  __syncthreads();

  const int wave = t >> 5;
  const int lane = t & 31;
  const int hh = lane >> 4;   // half-wave: selects K offset {0,2}
  const int ln = lane & 15;   // M index for A, N index for B/D
  const int col0 = wave * 16;

  v8f acc = {0.f, 0.f, 0.f, 0.f, 0.f, 0.f, 0.f, 0.f};

#pragma unroll
  for (int k0 = 0; k0 < IN_DIM; k0 += 4) {
    // A 16x4 fp32 layout: lanes 0-15 hold K={k0,k0+1}, lanes 16-31 K={k0+2,k0+3}
    v2f a, b;
    a.x = xt[ln * 132 + k0 + 2 * hh + 0];
    a.y = xt[ln * 132 + k0 + 2 * hh + 1];
    // B 4x16 fp32 layout (mirror of C/D): VGPR0 -> K = k0+2*hh, VGPR1 -> K+1
    b.x = W[(size_t)(k0 + 2 * hh + 0) * IN_DIM + col0 + ln];
    b.y = W[(size_t)(k0 + 2 * hh + 1) * IN_DIM + col0 + ln];
    acc = __builtin_amdgcn_wmma_f32_16x16x4_f32(
        /*neg_a=*/false, a, /*neg_b=*/false, b,
        /*c_mod=*/(short)0, acc, /*reuse_a=*/false, /*reuse_b=*/false);
  }

  // D layout: VGPR r -> M = r + 8*hh, N = ln
#pragma unroll
  for (int r = 0; r < 8; ++r) {
    int row = row0 + r + 8 * hh;
    if (row < N) O[(size_t)row * IN_DIM + col0 + ln] = acc[r] + bb[col0 + ln];
  }
}

// ---------------------------------------------------------------- 3. edge logits + segment max
// One wave per edge: lanes 0-15 = head 0, lanes 16-31 = head 1; each lane a float4.
__global__ __launch_bounds__(256) void gnn_edge_logits(
    const int* __restrict__ ei, int E,
    const float* __restrict__ Q, const float* __restrict__ K,
    float* __restrict__ logits, float* __restrict__ segmax) {
  int e = blockIdx.x * (blockDim.x >> 5) + (threadIdx.x >> 5);
  if (e >= E) return;  // wave-uniform exit
  int lane = threadIdx.x & 31;
  int h = lane >> 4, sub = lane & 15;
  int src = ei[e];
  int dst = ei[E + e];

  const float4 qv = *(const float4*)(Q + (size_t)dst * IN_DIM + h * 64 + sub * 4);
  const float4 kv = *(const float4*)(K + (size_t)src * IN_DIM + h * 64 + sub * 4);
  float p = qv.x * kv.x + qv.y * kv.y + qv.z * kv.z + qv.w * kv.w;
#pragma unroll
  for (int off = 1; off < 16; off <<= 1) p += __shfl_xor(p, off, 32);

  if (sub == 0) {
    float lg = p * 0.125f;  // 1/sqrt(64)
    logits[(size_t)e * 2 + h] = lg;
    atomicMaxFloat(&segmax[(size_t)dst * 2 + h], lg);
  }
}

// ---------------------------------------------------------------- 4. exp + denom
__global__ void gnn_edge_exp(const int* __restrict__ ei, int E,
                             float* __restrict__ logits,   // in-place -> ex
                             const float* __restrict__ segmax,
                             float* __restrict__ denom) {
  int tid = blockIdx.x * blockDim.x + threadIdx.x;
  if (tid >= 2 * E) return;
  int e = tid >> 1, h = tid & 1;
  int dst = ei[E + e];
  float ex = expf(logits[tid] - segmax[(size_t)dst * 2 + h]);
  logits[tid] = ex;
  atomicAdd(&denom[(size_t)dst * 2 + h], ex);
}

// ---------------------------------------------------------------- 5. weighted scatter
__global__ __launch_bounds__(256) void gnn_scatter(
    const int* __restrict__ ei, int E,
    const float* __restrict__ V, const float* __restrict__ ex,
    const float* __restrict__ denom, float* __restrict__ ACC) {
  int e = blockIdx.x * (blockDim.x >> 5) + (threadIdx.x >> 5);
  if (e >= E) return;
  int lane = threadIdx.x & 31;
  int h = lane >> 4, sub = lane & 15;
  int src = ei[e];
  int dst = ei[E + e];

  float a = ex[(size_t)e * 2 + h];
  float d = denom[(size_t)dst * 2 + h];
  a = a / fmaxf(d, 1e-16f);

  const float4 vv = *(const float4*)(V + (size_t)src * IN_DIM + h * 64 + sub * 4);
  float* o = ACC + (size_t)dst * IN_DIM + h * 64 + sub * 4;
  atomicAdd(o + 0, a * vv.x);
  atomicAdd(o + 1, a * vv.y);
  atomicAdd(o + 2, a * vv.z);
  atomicAdd(o + 3, a * vv.w);
}

// ---------------------------------------------------------------- 6. column stats
__global__ __launch_bounds__(128) void gnn_stats(const float* __restrict__ ACC, int N,
                                                 float* __restrict__ sums,
                                                 float* __restrict__ sumsq) {
  int col = threadIdx.x;  // 0..127, coalesced row sweeps
  float s = 0.0f, q = 0.0f;
  for (int r = blockIdx.x; r < N; r += gridDim.x) {
    float v = ACC[(size_t)r * IN_DIM + col];
    s += v;
    q += v * v;
  }
  atomicAdd(&sums[col], s);
  atomicAdd(&sumsq[col], q);
}

// ---------------------------------------------------------------- 7. BN + Mish
__global__ void gnn_bn_mish(const float* __restrict__ ACC, int N,
                            const float* __restrict__ sums, const float* __restrict__ sumsq,
                            const float* __restrict__ gamma, const float* __restrict__ beta,
                            float* __restrict__ out) {
  size_t tid = (size_t)blockIdx.x * blockDim.x + threadIdx.x;
  size_t total = (size_t)N * IN_DIM;
  if (tid >= total) return;
  int col = (int)(tid & (IN_DIM - 1));
  float invN = 1.0f / (float)N;
  float mean = sums[col] * invN;
  float var = sumsq[col] * invN - mean * mean;  // biased var (ddof=0)
  float y = gamma[col] * (ACC[tid] - mean) * rsqrtf(var + EPS_BN) + beta[col];
  float sp = (y > 20.0f) ? y : log1pf(expf(y));  // softplus
  out[tid] = y * tanhf(sp);                      // Mish
}

// ---------------------------------------------------------------- launch
extern "C" void kernel_launch(void* const* d_in, const int* in_sizes, int n_in,
                              void* d_out, int out_size, void* d_ws, size_t ws_size,
                              hipStream_t stream) {
  const float* x  = (const float*)d_in[0];
  const int*   ei = (const int*)d_in[1];
  const float* Wq = (const float*)d_in[2];
  const float* bq = (const float*)d_in[3];
  const float* Wk = (const float*)d_in[4];
  const float* bk = (const float*)d_in[5];
  const float* Wv = (const float*)d_in[6];
  const float* bv = (const float*)d_in[7];
  const float* Ws = (const float*)d_in[8];
  const float* bs = (const float*)d_in[9];
  const float* gamma = (const float*)d_in[10];
  const float* beta  = (const float*)d_in[11];

  const int N = in_sizes[0] / IN_DIM;
  const int E = in_sizes[1] / 2;

  float* ws = (float*)d_ws;
  size_t o = 0;
  float* Q   = ws + o; o += (size_t)N * IN_DIM;
  float* K   = ws + o; o += (size_t)N * IN_DIM;
  float* V   = ws + o; o += (size_t)N * IN_DIM;
  float* ACC = ws + o; o += (size_t)N * IN_DIM;
  float* EX  = ws + o; o += (size_t)E * 2;   // logits, then ex in-place
  float* SM  = ws + o; o += (size_t)N * 2;
  float* DN  = ws + o; o += (size_t)N * 2;
  float* SUM = ws + o; o += IN_DIM;
  float* SSQ = ws + o; o += IN_DIM;

  const int NH = N * 2;
  gnn_init<<<(NH + 255) / 256, 256, 0, stream>>>(SM, DN, NH, SUM, SSQ);

  dim3 gemm_grid((N + 15) / 16, 4);
  gnn_gemm_qkvs<<<gemm_grid, 256, 0, stream>>>(x, N, Wq, bq, Wk, bk, Wv, bv, Ws, bs,
                                               Q, K, V, ACC);

  gnn_edge_logits<<<(E + 7) / 8, 256, 0, stream>>>(ei, E, Q, K, EX, SM);
  gnn_edge_exp<<<(2 * E + 255) / 256, 256, 0, stream>>>(ei, E, EX, SM, DN);
  gnn_scatter<<<(E + 7) / 8, 256, 0, stream>>>(ei, E, V, EX, DN, ACC);

  gnn_stats<<<512, 128, 0, stream>>>(ACC, N, SUM, SSQ);

  size_t total = (size_t)N * IN_DIM;
  gnn_bn_mish<<<(unsigned)((total + 255) / 256), 256, 0, stream>>>(ACC, N, SUM, SSQ,
                                                                   gamma, beta,
                                                                   (float*)d_out);
}